// ToposMambaBlock_5257039970326
// MI455X (gfx1250) — compile-verified
//
#include <hip/hip_runtime.h>
#include <hip/hip_bf16.h>
#include <math.h>

// Problem sizes (fixed by reference setup_inputs)
#define BB 16
#define SS 1024
#define DD 1024
#define NN 64
#define MM (BB * SS)   // 16384 rows for the Bt GEMM

typedef float v2f __attribute__((ext_vector_type(2)));
typedef float v4f __attribute__((ext_vector_type(4)));
typedef float v8f __attribute__((ext_vector_type(8)));

__device__ __forceinline__ float sig_f(float z) {
    return 1.0f / (1.0f + __expf(-z));
}
__device__ __forceinline__ float clamp01(float v) {
    return __builtin_amdgcn_fmed3f(v, 0.0f, 1.0f);   // v_med3_f32
}

// Packed f32 VOP3P ops (CDNA5 §15.10: V_PK_FMA_F32 op 31, V_PK_MUL_F32 op 40).
// 64-bit VGPR-pair operands; VALU->VALU hazards are hardware-interlocked, so
// opaque inline asm needs no manual waits.
__device__ __forceinline__ v2f pk_fma(v2f a, v2f b, v2f c) {
    v2f d;
    asm("v_pk_fma_f32 %0, %1, %2, %3" : "=v"(d) : "v"(a), "v"(b), "v"(c));
    return d;
}
__device__ __forceinline__ v2f pk_mul(v2f a, v2f b) {
    v2f d;
    asm("v_pk_mul_f32 %0, %1, %2" : "=v"(d) : "v"(a), "v"(b));
    return d;
}

// ---------------------------------------------------------------------------
// Kernel 1: precompute sigmoid-transformed weights into workspace.
//   BwI: sigmoid(W_B) transposed AND K-pair interleaved for b64 B-frag loads:
//        BwI[((k>>1)*64 + n)*2 + (k&1)] = sigmoid(W_B[n*D + k])
//   Asig[d*64+n] = sigmoid(A[d*64+n])
//   Csig[d*64+n] = sigmoid(W_C[d*64+n])
// ---------------------------------------------------------------------------
__global__ void topos_prep_kernel(const float* __restrict__ A,
                                  const float* __restrict__ W_B,
                                  const float* __restrict__ W_C,
                                  float* __restrict__ BwI,
                                  float* __restrict__ Asig,
                                  float* __restrict__ Csig) {
    int i = blockIdx.x * blockDim.x + threadIdx.x;   // 0 .. D*N-1
    if (i >= DD * NN) return;
    int k = i >> 6;          // 0..1023 (the D/K index)
    int n = i & 63;
    Asig[i] = sig_f(A[i]);
    Csig[i] = sig_f(W_C[i]);
    BwI[(((k >> 1) * NN + n) << 1) | (k & 1)] = sig_f(W_B[n * DD + k]);
}

// ---------------------------------------------------------------------------
// Kernel 2: Bt[M, N] = X[M, D] * sigmoid(W_B)^T using V_WMMA_F32_16X16X4_F32.
// One wave32 per 16-row M-tile, 4 accumulators cover N=64 (4 tiles of 16).
// A-frag (16x4 f32): lanes 0-15 hold K=0,1 ; lanes 16-31 hold K=2,3.
// B-frag (4x16 f32): lanes 0-15 -> N, vgpr0=K0 vgpr1=K1; lanes 16-31 -> K2,K3.
// C/D (16x16 f32): vgpr i = row i (lanes 0-15) / row i+8 (lanes 16-31).
// B fragments come from the pair-interleaved BwI as single b64 loads.
// ---------------------------------------------------------------------------
__global__ __launch_bounds__(256)
void topos_gemm_bt_kernel(const float* __restrict__ X,
                          const float* __restrict__ BwI,
                          float* __restrict__ Bt) {
    const int lane   = threadIdx.x & 31;
    const int wave   = threadIdx.x >> 5;
    const int mtile  = blockIdx.x * 8 + wave;        // 0 .. MM/16 - 1
    const int M0     = mtile * 16;
    const int half   = lane >> 4;                    // 0: K pair 0/1, 1: K pair 2/3
    const int mr     = lane & 15;                    // row within tile (A side)
    const int nc     = lane & 15;                    // col within tile (B side)

    const float* __restrict__ arow = X + (size_t)(M0 + mr) * DD;

    v8f acc0 = {}, acc1 = {}, acc2 = {}, acc3 = {};

    for (int k0 = 0; k0 < DD; k0 += 4) {
        const int ka = k0 + 2 * half;                // even
        v2f af = *(const v2f*)(arow + ka);

        // pair-interleaved: element pair (ka, ka+1) at ((ka>>1)*64 + n) * 2
        const float* __restrict__ bp = BwI + (size_t)(((ka >> 1) * NN + nc) << 1);
        v2f b0 = *(const v2f*)(bp + 0);
        v2f b1 = *(const v2f*)(bp + 32);
        v2f b2 = *(const v2f*)(bp + 64);
        v2f b3 = *(const v2f*)(bp + 96);

        acc0 = __builtin_amdgcn_wmma_f32_16x16x4_f32(false, af, false, b0,
                                                     (short)0, acc0, false, false);
        acc1 = __builtin_amdgcn_wmma_f32_16x16x4_f32(false, af, false, b1,
                                                     (short)0, acc1, false, false);
        acc2 = __builtin_amdgcn_wmma_f32_16x16x4_f32(false, af, false, b2,
                                                     (short)0, acc2, false, false);
        acc3 = __builtin_amdgcn_wmma_f32_16x16x4_f32(false, af, false, b3,
                                                     (short)0, acc3, false, false);
    }

#pragma unroll
    for (int i = 0; i < 8; ++i) {
        const int m = i + 8 * half;
        float* p = Bt + (size_t)(M0 + m) * NN + nc;
        p[0]  = acc0[i];
        p[16] = acc1[i];
        p[32] = acc2[i];
        p[48] = acc3[i];
    }
}

// ---------------------------------------------------------------------------
// Kernel 3: sequential scan. One workgroup per batch, thread = one d.
// h[d, 0..63] lives in VGPRs (32 x f2); Asig/Csig streamed from L2; bt row
// is a same-address broadcast load. Inner math uses forced V_PK_FMA_F32 /
// V_PK_MUL_F32 (2 f32 FLOPs per VALU issue), clamp via v_med3_f32.
// Per-step layernorm over D via wave shuffles + LDS tree, two barriers.
// ---------------------------------------------------------------------------
__global__ __launch_bounds__(1024, 1)
void topos_scan_kernel(const float* __restrict__ x,
                       const float* __restrict__ gamma,
                       const float* __restrict__ Asig,
                       const float* __restrict__ Csig,
                       const float* __restrict__ Bt,
                       float* __restrict__ out) {
    const int b = blockIdx.x;       // batch
    const int d = threadIdx.x;      // 0..1023
    const int wave = d >> 5;
    const int lane = d & 31;

    __shared__ float red_s[32];
    __shared__ float red_q[32];
    __shared__ float bcast[2];      // mu, inv_sigma

    const float gd = sig_f(gamma[d]);

    const v4f* __restrict__ a4 = (const v4f*)(Asig + (size_t)d * NN);
    const v4f* __restrict__ c4 = (const v4f*)(Csig + (size_t)d * NN);
    const float* __restrict__ xb = x  + (size_t)b * SS * DD;
    const float* __restrict__ bb = Bt + (size_t)b * SS * NN;
    float* __restrict__ ob       = out + (size_t)b * SS * DD;

    v2f h2[NN / 2];
#pragma unroll
    for (int p = 0; p < NN / 2; ++p) h2[p] = (v2f){0.0f, 0.0f};

    for (int t = 0; t < SS; ++t) {
        const float xv = xb[(size_t)t * DD + d];
        const v4f* __restrict__ bt4 = (const v4f*)(bb + (size_t)t * NN);

        if (t + 1 < SS) {
            __builtin_prefetch(xb + (size_t)(t + 1) * DD + d, 0, 0);
            __builtin_prefetch(bb + (size_t)(t + 1) * NN, 0, 0);
        }

        const v2f xv2 = (v2f){xv, xv};
        v2f y2a = (v2f){0.0f, 0.0f};     // two accumulators: shorter dep chains
        v2f y2b = (v2f){0.0f, 0.0f};
#pragma unroll
        for (int q = 0; q < 16; ++q) {
            const v4f av = a4[q];
            const v4f cv = c4[q];
            const v4f bv = bt4[q];
            const v2f alo = __builtin_shufflevector(av, av, 0, 1);
            const v2f ahi = __builtin_shufflevector(av, av, 2, 3);
            const v2f clo = __builtin_shufflevector(cv, cv, 0, 1);
            const v2f chi = __builtin_shufflevector(cv, cv, 2, 3);
            const v2f blo = __builtin_shufflevector(bv, bv, 0, 1);
            const v2f bhi = __builtin_shufflevector(bv, bv, 2, 3);

            v2f hv = pk_fma(h2[2 * q + 0], alo, pk_mul(blo, xv2));
            hv.x = clamp01(hv.x);
            hv.y = clamp01(hv.y);
            h2[2 * q + 0] = hv;
            y2a = pk_fma(hv, clo, y2a);

            hv = pk_fma(h2[2 * q + 1], ahi, pk_mul(bhi, xv2));
            hv.x = clamp01(hv.x);
            hv.y = clamp01(hv.y);
            h2[2 * q + 1] = hv;
            y2b = pk_fma(hv, chi, y2b);
        }
        const float y = (y2a.x + y2a.y) + (y2b.x + y2b.y);

        // workgroup reduction of sum(y) and sum(y*y) over D=1024
        float s = y, sq = y * y;
#pragma unroll
        for (int off = 16; off > 0; off >>= 1) {
            s  += __shfl_down(s,  off, 32);
            sq += __shfl_down(sq, off, 32);
        }
        if (lane == 0) { red_s[wave] = s; red_q[wave] = sq; }
        __syncthreads();
        if (wave == 0) {
            float ps = red_s[lane], pq = red_q[lane];
#pragma unroll
            for (int off = 16; off > 0; off >>= 1) {
                ps += __shfl_down(ps, off, 32);
                pq += __shfl_down(pq, off, 32);
            }
            if (lane == 0) {
                const float mu  = ps * (1.0f / (float)DD);
                const float var = pq * (1.0f / (float)DD) - mu * mu;
                bcast[0] = mu;
                bcast[1] = __frsqrt_rn(var + 1e-5f);
            }
        }
        __syncthreads();

        const float yn = (y - bcast[0]) * bcast[1] * gd;
        ob[(size_t)t * DD + d] = clamp01(yn + xv);
    }
}

// ---------------------------------------------------------------------------
// Host-side launcher
// ---------------------------------------------------------------------------
extern "C" void kernel_launch(void* const* d_in, const int* in_sizes, int n_in,
                              void* d_out, int out_size, void* d_ws, size_t ws_size,
                              hipStream_t stream) {
    const float* x     = (const float*)d_in[0];   // [B, S, D]
    const float* A     = (const float*)d_in[1];   // [D, N]
    const float* W_B   = (const float*)d_in[2];   // [N, D]
    const float* W_C   = (const float*)d_in[3];   // [D, N]
    const float* gamma = (const float*)d_in[4];   // [D]
    float* out = (float*)d_out;                   // [B, S, D]

    float* ws   = (float*)d_ws;
    float* BwI  = ws;                       // [D/2, N, 2]  65536 floats
    float* Asig = ws + DD * NN;             // [D, N]
    float* Csig = ws + 2 * DD * NN;         // [D, N]
    float* Bt   = ws + 3 * DD * NN;         // [M, N]       1048576 floats

    // 1) sigmoid precompute (65536 elements)
    topos_prep_kernel<<<(DD * NN + 255) / 256, 256, 0, stream>>>(
        A, W_B, W_C, BwI, Asig, Csig);

    // 2) Bt = X * sigmoid(W_B)^T via f32 WMMA. 1024 M-tiles, 8 waves/block.
    topos_gemm_bt_kernel<<<(MM / 16) / 8, 256, 0, stream>>>(x, BwI, Bt);

    // 3) sequential scan, one workgroup per batch element
    topos_scan_kernel<<<BB, 1024, 0, stream>>>(x, gamma, Asig, Csig, Bt, out);
}